// TemporalCore_18141941858449
// MI455X (gfx1250) — compile-verified
//
#include <hip/hip_runtime.h>
#include <math.h>

#define L_SEQ 512
#define BSZ   4
#define DIM   512
#define NST   512
#define M_ROWS (L_SEQ * BSZ)   // 2048 rows for both GEMMs

typedef __attribute__((ext_vector_type(2))) float v2f;
typedef __attribute__((ext_vector_type(8))) float v8f;

__device__ __forceinline__ float softplus_f(float x) {
    // stable softplus with fast-path log/exp (v_log_f32 / v_exp_f32):
    // max(x,0) + log(1 + exp(-|x|)). When exp(-|x|) underflows toward 0 the
    // log term is negligible relative to max(x,0), so fast log is sufficient.
    float e = __expf(-fabsf(x));
    return fmaxf(x, 0.0f) + __logf(1.0f + e);
}

// ---------------------------------------------------------------------------
// Kernel 1: delta = softplus(X @ W^T + b), u = delta * X
//   X: (M=2048, K=512) row-major;  W: (N=512, K=512) row-major (W[e][d])
// One wave computes a 16x64 strip: 1 M-tile x 4 N-tiles. The A fragment is
// loaded once per k-step and reused by 4 V_WMMA_F32_16X16X4_F32 ops.
// f32 WMMA fragment layout (ISA 7.12.2):
//   A 16x4:  lanes 0-15 -> M=lane, VGPR0=K0,VGPR1=K1 ; lanes 16-31 -> K2,K3
//   B 4x16:  lanes 0-15 -> N=lane, VGPR0=K0,VGPR1=K1 ; lanes 16-31 -> K2,K3
//   C/D:     VGPR r: lanes 0-15 -> M=r, N=lane ; lanes 16-31 -> M=r+8
// ---------------------------------------------------------------------------
__global__ void __launch_bounds__(256)
delta_gemm_kernel(const float* __restrict__ X,
                  const float* __restrict__ W,
                  const float* __restrict__ bias,
                  float* __restrict__ delta_out,
                  float* __restrict__ u_out)
{
    const int lane = threadIdx.x & 31;
    const int wave = threadIdx.x >> 5;
    const int strip = blockIdx.x * 8 + wave;       // 128 M-tiles * 8 N-strips
    const int n_strips = NST / 64;                 // 8
    const int tm  = (strip / n_strips) * 16;
    const int tn0 = (strip % n_strips) * 64;

    const int half = lane >> 4;                    // 0: K0/K1, 1: K2/K3
    const int l16  = lane & 15;

    const float* xrow = X + (size_t)(tm + l16) * DIM;
    const float* wrow = W + (size_t)(tn0 + l16) * DIM;   // n-tile stride 16*DIM

    v8f acc0 = {}, acc1 = {}, acc2 = {}, acc3 = {};
    #pragma unroll 2
    for (int k0 = 0; k0 < DIM; k0 += 4) {
        const int ka = k0 + half * 2;
        v2f a  = *(const v2f*)(xrow + ka);
        v2f b0 = *(const v2f*)(wrow + ka);
        v2f b1 = *(const v2f*)(wrow + 16 * DIM + ka);
        v2f b2 = *(const v2f*)(wrow + 32 * DIM + ka);
        v2f b3 = *(const v2f*)(wrow + 48 * DIM + ka);
        acc0 = __builtin_amdgcn_wmma_f32_16x16x4_f32(false, a, false, b0, (short)0, acc0, false, false);
        acc1 = __builtin_amdgcn_wmma_f32_16x16x4_f32(false, a, false, b1, (short)0, acc1, false, false);
        acc2 = __builtin_amdgcn_wmma_f32_16x16x4_f32(false, a, false, b2, (short)0, acc2, false, false);
        acc3 = __builtin_amdgcn_wmma_f32_16x16x4_f32(false, a, false, b3, (short)0, acc3, false, false);
    }

    v8f accs[4] = {acc0, acc1, acc2, acc3};
    #pragma unroll
    for (int nt = 0; nt < 4; ++nt) {
        const int n  = tn0 + nt * 16 + l16;
        const float bn = bias[n];
        #pragma unroll
        for (int r = 0; r < 8; ++r) {
            const int m   = tm + half * 8 + r;
            const int idx = m * NST + n;
            const float d = softplus_f(accs[nt][r] + bn);
            delta_out[idx] = d;
            u_out[idx]     = d * X[idx];           // NST == DIM: indices align
        }
    }
}

// ---------------------------------------------------------------------------
// Kernel 2: selective-scan. One wave32 per (b,d) chain; each lane owns 16 of
// the 512 state elements (n = lane + 32*j). h stays entirely in registers.
//   per step: h = exp(delta_t * A[d,n]) * h + u_t * Bm[d,n];  y_t = sum_n h
// ---------------------------------------------------------------------------
__global__ void __launch_bounds__(256)
scan_kernel(const float* __restrict__ A,       // (DIM, NST)
            const float* __restrict__ Bm,      // (DIM, NST)
            const float* __restrict__ delta,   // (L,B,D)
            const float* __restrict__ u,       // (L,B,D)
            float* __restrict__ y)             // (L,B,D)
{
    const int lane = threadIdx.x & 31;
    const int wave = threadIdx.x >> 5;
    const int p = blockIdx.x * 8 + wave;       // 0 .. B*DIM-1
    const int b = p / DIM;
    const int d = p % DIM;

    float Ar[16], Br[16], h[16];
    #pragma unroll
    for (int j = 0; j < 16; ++j) {
        const int n = lane + 32 * j;           // coalesced across the wave
        Ar[j] = A [d * NST + n];
        Br[j] = Bm[d * NST + n];
        h[j]  = 0.0f;
    }

    const int stride = BSZ * DIM;
    int base = b * DIM + d;
    for (int t = 0; t < L_SEQ; ++t) {
        const float dt = delta[base];          // wave-uniform scalar
        const float ut = u[base];
        float acc = 0.0f;
        #pragma unroll
        for (int j = 0; j < 16; ++j) {
            h[j] = __expf(dt * Ar[j]) * h[j] + ut * Br[j];   // v_exp_f32 + FMA
            acc += h[j];
        }
        #pragma unroll
        for (int off = 16; off >= 1; off >>= 1)
            acc += __shfl_xor(acc, off, 32);   // wave32 reduction
        if (lane == 0) y[base] = acc;
        base += stride;
    }
}

// ---------------------------------------------------------------------------
// Kernel 3: out = Y @ C + X * D_param
//   Y: (M=2048, K=512);  C: (K=512, N=512) row-major (K-major already)
// Same 16x64-strip WMMA tiling; B fragments gathered with stride NST.
// ---------------------------------------------------------------------------
__global__ void __launch_bounds__(256)
out_gemm_kernel(const float* __restrict__ Y,
                const float* __restrict__ C,
                const float* __restrict__ X,
                const float* __restrict__ Dp,
                float* __restrict__ out)
{
    const int lane = threadIdx.x & 31;
    const int wave = threadIdx.x >> 5;
    const int strip = blockIdx.x * 8 + wave;
    const int n_strips = NST / 64;
    const int tm  = (strip / n_strips) * 16;
    const int tn0 = (strip % n_strips) * 64;

    const int half = lane >> 4;
    const int l16  = lane & 15;

    const float* yrow = Y + (size_t)(tm + l16) * DIM;
    const float* ccol = C + (tn0 + l16);       // column base, row stride NST

    v8f acc0 = {}, acc1 = {}, acc2 = {}, acc3 = {};
    #pragma unroll 2
    for (int k0 = 0; k0 < DIM; k0 += 4) {
        const int ka = k0 + half * 2;
        v2f a = *(const v2f*)(yrow + ka);
        const float* c0 = ccol + (size_t)ka * NST;
        v2f b0, b1, b2, b3;
        b0.x = c0[0];        b0.y = c0[NST];
        b1.x = c0[16];       b1.y = c0[NST + 16];
        b2.x = c0[32];       b2.y = c0[NST + 32];
        b3.x = c0[48];       b3.y = c0[NST + 48];
        acc0 = __builtin_amdgcn_wmma_f32_16x16x4_f32(false, a, false, b0, (short)0, acc0, false, false);
        acc1 = __builtin_amdgcn_wmma_f32_16x16x4_f32(false, a, false, b1, (short)0, acc1, false, false);
        acc2 = __builtin_amdgcn_wmma_f32_16x16x4_f32(false, a, false, b2, (short)0, acc2, false, false);
        acc3 = __builtin_amdgcn_wmma_f32_16x16x4_f32(false, a, false, b3, (short)0, acc3, false, false);
    }

    v8f accs[4] = {acc0, acc1, acc2, acc3};
    #pragma unroll
    for (int nt = 0; nt < 4; ++nt) {
        const int n   = tn0 + nt * 16 + l16;
        const float dpn = Dp[n];
        #pragma unroll
        for (int r = 0; r < 8; ++r) {
            const int m   = tm + half * 8 + r;
            const int idx = m * NST + n;
            out[idx] = accs[nt][r] + X[idx] * dpn;
        }
    }
}

// ---------------------------------------------------------------------------
extern "C" void kernel_launch(void* const* d_in, const int* in_sizes, int n_in,
                              void* d_out, int out_size, void* d_ws, size_t ws_size,
                              hipStream_t stream) {
    const float* x_seq = (const float*)d_in[0];   // (L,B,D)
    const float* A     = (const float*)d_in[1];   // (D,N)
    const float* Bm    = (const float*)d_in[2];   // (D,N)
    const float* C     = (const float*)d_in[3];   // (D,N)
    const float* Dp    = (const float*)d_in[4];   // (1,D)
    const float* W     = (const float*)d_in[5];   // (D,D)
    const float* bd    = (const float*)d_in[6];   // (D)
    float* out = (float*)d_out;

    float* delta = (float*)d_ws;                  // 4 MB
    float* u     = delta + (size_t)M_ROWS * DIM;  // 4 MB
    float* y     = u     + (size_t)M_ROWS * DIM;  // 4 MB

    const dim3 blk(256);
    // 128 M-tiles * 8 N-strips = 1024 wave-strips, 8 waves/block -> 128 blocks
    const int strips = (M_ROWS / 16) * (NST / 64);
    const int blocks = strips / 8;

    delta_gemm_kernel<<<blocks, blk, 0, stream>>>(x_seq, W, bd, delta, u);
    scan_kernel<<<(BSZ * DIM) / 8, blk, 0, stream>>>(A, Bm, delta, u, y);
    out_gemm_kernel<<<blocks, blk, 0, stream>>>(y, C, x_seq, Dp, out);
}